// DecoderRNN_76373108457983
// MI455X (gfx1250) — compile-verified
//
#include <hip/hip_runtime.h>
#include <math.h>

// ---- problem sizes ----
#define BB 64
#define TT 32
#define LL 49
#define FF 2048
#define CC 128
#define HH 1024
#define EE 512
#define VV 12000
#define XK (EE + FF + CC) /* 2688 */

#define KC 128          /* K-chunk staged in LDS by the TDM */
#define KCP (KC + 2)    /* TDM pads 2 DWORDs every 128 -> bank-conflict-free rows */

typedef __attribute__((ext_vector_type(2))) float v2f;
typedef __attribute__((ext_vector_type(8))) float v8f;
typedef __attribute__((ext_vector_type(4))) unsigned int v4u;
typedef __attribute__((ext_vector_type(4))) int v4i;
typedef __attribute__((ext_vector_type(8))) int v8i;

// fp32 WMMA: D(16x16) += A(16x4) * B(4x16), full fp32 precision (matches reference dtype)
__device__ __forceinline__ v8f wmma4(v2f a, v2f b, v8f c) {
  return __builtin_amdgcn_wmma_f32_16x16x4_f32(
      /*neg_a=*/false, a, /*neg_b=*/false, b, /*c_mod=*/(short)0, c,
      /*reuse_a=*/false, /*reuse_b=*/false);
}

// ---- Tensor DMA descriptor packing (CDNA5 ISA ch.8, D# groups) ----
// Group 0: count=1 | lds_addr | global_addr(57b) | type=2
__device__ __forceinline__ v4u tdm_g0(unsigned lds_addr, unsigned long long gaddr) {
  v4u g;
  g.x = 1u;                                              // count=1, is_restore=0
  g.y = lds_addr;                                        // LDS byte address
  g.z = (unsigned)(gaddr & 0xFFFFFFFFu);                 // global_addr[31:0]
  g.w = (unsigned)((gaddr >> 32) & 0x01FFFFFFu) | (2u << 30);  // [56:32] | type=2
  return g;
}
// Group 1: 2D tile, data_size=4B, LDS padding 2 DWORDs per 128 DWORDs.
__device__ __forceinline__ v8i tdm_g1_2d(unsigned tensor_d0, unsigned tensor_d1,
                                         unsigned tile_d0, unsigned tile_d1,
                                         unsigned long long stride0) {
  v8i g;
  g[0] = (int)((2u << 16)        // data_size = 4 bytes
             | (1u << 20)        // pad_enable
             | (6u << 22)        // pad_interval: 2^(6+1)=128 DWORDs
             | (1u << 25));      // pad_amount: 1+1 = 2 DWORDs
  g[1] = (int)((tensor_d0 & 0xFFFFu) << 16);                         // dim0[15:0]
  g[2] = (int)(((tensor_d0 >> 16) & 0xFFFFu) | ((tensor_d1 & 0xFFFFu) << 16));
  g[3] = (int)(((tensor_d1 >> 16) & 0xFFFFu) | ((tile_d0 & 0xFFFFu) << 16));
  g[4] = (int)(tile_d1 & 0xFFFFu);                                   // tile_dim1, tile_dim2=0
  g[5] = (int)(stride0 & 0xFFFFFFFFull);                             // dim0_stride[31:0]
  g[6] = (int)((stride0 >> 32) & 0xFFFFull);                         // dim0_stride[47:32]
  g[7] = 0;
  return g;
}

// Stage A[0:64, kc:kc+KC) (row stride ldk floats) into LDS with padded rows.
// Wave-level op: TDM ignores EXEC; caller gates to one wave per block.
__device__ __forceinline__ void tdm_stage_panel(const float* A, int ldk, int kc,
                                                float* sA) {
  unsigned lds = (unsigned)(unsigned long long)(size_t)sA;
  unsigned long long gaddr = (unsigned long long)(size_t)(A + kc);
  v4u g0 = tdm_g0(lds, gaddr);
  v8i g1 = tdm_g1_2d((unsigned)ldk, BB, KC, BB, (unsigned long long)ldk);
  v4i z4 = {};
  v8i z8 = {};
  __builtin_amdgcn_tensor_load_to_lds(g0, g1, z4, z4, z8, 0);
  __builtin_amdgcn_s_wait_tensorcnt(0);
}

// ---------------- setup kernels ----------------

__global__ void k_mean_feat(const float* __restrict__ features, float* __restrict__ meanf) {
  int idx = blockIdx.x * blockDim.x + threadIdx.x;
  if (idx >= BB * FF) return;
  int b = idx / FF, f = idx % FF;
  const float* p = features + (size_t)b * LL * FF + f;
  float s = 0.f;
  for (int l = 0; l < LL; ++l) s += p[(size_t)l * FF];
  meanf[idx] = s * (1.0f / LL);
}

__global__ void k_embed(const float* __restrict__ emb, const int* __restrict__ captions,
                        float* __restrict__ embed) {
  int idx = blockIdx.x * blockDim.x + threadIdx.x;
  if (idx >= BB * TT * EE) return;
  int e = idx % EE;
  int bt = idx / EE;
  int tok = captions[bt];
  embed[idx] = emb[(size_t)tok * EE + e];
}

// h0 = meanf @ Wh0^T + bh0 ; c0 = meanf @ Wc0^T + bc0   (M=64,N=1024,K=2048)
__global__ void __launch_bounds__(256) k_init_hc(
    const float* __restrict__ meanf, const float* __restrict__ Wh0,
    const float* __restrict__ bh0, const float* __restrict__ Wc0,
    const float* __restrict__ bc0, float* __restrict__ h0, float* __restrict__ c0) {
  const int tilesN = HH / 16;
  int wave = (blockIdx.x * blockDim.x + threadIdx.x) >> 5;
  if (wave >= (BB / 16) * tilesN) return;
  int lane = threadIdx.x & 31, half = lane >> 4, l16 = lane & 15;
  int mt = wave / tilesN, nt = wave % tilesN;
  int m = mt * 16 + l16, n = nt * 16 + l16;
  v8f accH = {}, accC = {};
  const float* ap  = meanf + (size_t)m * FF;
  const float* bhp = Wh0   + (size_t)n * FF;
  const float* bcp = Wc0   + (size_t)n * FF;
  for (int k = 0; k < FF; k += 4) {
    int kk = k + 2 * half;
    v2f a  = *(const v2f*)(ap + kk);
    v2f bh = *(const v2f*)(bhp + kk);
    v2f bc = *(const v2f*)(bcp + kk);
    accH = wmma4(a, bh, accH);
    accC = wmma4(a, bc, accC);
  }
  float bh_ = bh0[n], bc_ = bc0[n];
  for (int r = 0; r < 8; ++r) {
    int mm = mt * 16 + r + 8 * half;
    h0[(size_t)mm * HH + n] = accH[r] + bh_;
    c0[(size_t)mm * HH + n] = accC[r] + bc_;
  }
}

// feat_proj = features(BL,F) @ att_W(F,H)   (M=3136,N=1024,K=2048) — one-time
__global__ void __launch_bounds__(256) k_featproj(
    const float* __restrict__ features, const float* __restrict__ att_W,
    float* __restrict__ feat_proj) {
  const int tilesN = HH / 16, tilesM = (BB * LL) / 16;
  int wave = (blockIdx.x * blockDim.x + threadIdx.x) >> 5;
  if (wave >= tilesM * tilesN) return;
  int lane = threadIdx.x & 31, half = lane >> 4, l16 = lane & 15;
  int mt = wave / tilesN, nt = wave % tilesN;
  int m = mt * 16 + l16, n = nt * 16 + l16;
  v8f acc = {};
  const float* ap = features + (size_t)m * FF;
  const float* bp = att_W + n;
  for (int k = 0; k < FF; k += 4) {
    int kk = k + 2 * half;
    v2f a = *(const v2f*)(ap + kk);
    v2f b;
    b.x = bp[(size_t)kk * HH];
    b.y = bp[(size_t)(kk + 1) * HH];
    acc = wmma4(a, b, acc);
  }
  for (int r = 0; r < 8; ++r) {
    int mm = mt * 16 + r + 8 * half;
    feat_proj[(size_t)mm * HH + n] = acc[r];
  }
}

// ---------------- per-step kernels ----------------

// hU = h @ att_U   (M=64,N=1024,K=1024), att_U row-major KxN. Small: direct global.
__global__ void __launch_bounds__(256) k_hU(
    const float* __restrict__ h, const float* __restrict__ att_U, float* __restrict__ hU) {
  const int tilesN = HH / 16;
  int wave = (blockIdx.x * blockDim.x + threadIdx.x) >> 5;
  if (wave >= (BB / 16) * tilesN) return;
  int lane = threadIdx.x & 31, half = lane >> 4, l16 = lane & 15;
  int mt = wave / tilesN, nt = wave % tilesN;
  int m = mt * 16 + l16, n = nt * 16 + l16;
  v8f acc = {};
  const float* ap = h + (size_t)m * HH;
  const float* bp = att_U + n;
  for (int k = 0; k < HH; k += 4) {
    int kk = k + 2 * half;
    v2f a = *(const v2f*)(ap + kk);
    v2f b;
    b.x = bp[(size_t)kk * HH];
    b.y = bp[(size_t)(kk + 1) * HH];
    acc = wmma4(a, b, acc);
  }
  for (int r = 0; r < 8; ++r) {
    int mm = mt * 16 + r + 8 * half;
    hU[(size_t)mm * HH + n] = acc[r];
  }
}

// scores -> softmax -> attention weights (to d_out) + context. One block per batch row.
__global__ void __launch_bounds__(256) k_attention(
    const float* __restrict__ feat_proj, const float* __restrict__ hU,
    const float* __restrict__ att_v, const float* __restrict__ features,
    float* __restrict__ w_out, float* __restrict__ context, int t) {
  __shared__ float s_score[LL];
  int b = blockIdx.x;
  int wid = threadIdx.x >> 5, lane = threadIdx.x & 31;
  const float* fp = feat_proj + (size_t)b * LL * HH;
  const float* hu = hU + (size_t)b * HH;
  for (int l = wid; l < LL; l += 8) {
    const float* fpl = fp + (size_t)l * HH;
    float acc = 0.f;
    for (int a = lane; a < HH; a += 32)
      acc += tanhf(fpl[a] + hu[a]) * att_v[a];
    for (int off = 16; off > 0; off >>= 1)
      acc += __shfl_down(acc, off, 32);
    if (lane == 0) s_score[l] = acc;
  }
  __syncthreads();
  if (threadIdx.x == 0) {
    float mx = s_score[0];
    for (int l = 1; l < LL; ++l) mx = fmaxf(mx, s_score[l]);
    float s = 0.f;
    for (int l = 0; l < LL; ++l) { float e = expf(s_score[l] - mx); s_score[l] = e; s += e; }
    float inv = 1.f / s;
    for (int l = 0; l < LL; ++l) s_score[l] *= inv;
  }
  __syncthreads();
  if (threadIdx.x < LL)
    w_out[((size_t)b * TT + t) * LL + threadIdx.x] = s_score[threadIdx.x];
  const float* fb = features + (size_t)b * LL * FF;
  for (int f = threadIdx.x; f < FF; f += 256) {
    float s = 0.f;
    for (int l = 0; l < LL; ++l) s += s_score[l] * fb[(size_t)l * FF + f];
    context[(size_t)b * FF + f] = s;
  }
}

// xcat = [embed_t | context | category_enc]  (64 x 2688)
__global__ void k_xcat(const float* __restrict__ embed, const float* __restrict__ context,
                       const float* __restrict__ cat, float* __restrict__ xcat, int t) {
  int idx = blockIdx.x * blockDim.x + threadIdx.x;
  if (idx >= BB * XK) return;
  int b = idx / XK, k = idx % XK;
  float v;
  if (k < EE)            v = embed[((size_t)b * TT + t) * EE + k];
  else if (k < EE + FF)  v = context[(size_t)b * FF + (k - EE)];
  else                   v = cat[(size_t)b * CC + (k - EE - FF)];
  xcat[idx] = v;
}

// gates = xcat @ W_ih^T + b_ih + h @ W_hh^T + b_hh   (M=64,N=4096)
// Each wave: all 4 M-tiles x 1 N-tile (B loaded once, 4 WMMAs).
// A panels TDM-staged into LDS, rows padded by 2 DWORDs (bank-conflict free).
__global__ void __launch_bounds__(256) k_gates(
    const float* __restrict__ xcat, const float* __restrict__ W_ih,
    const float* __restrict__ h, const float* __restrict__ W_hh,
    const float* __restrict__ b_ih, const float* __restrict__ b_hh,
    float* __restrict__ gates) {
  __shared__ float sA[BB * KCP];
  int waveInBlock = threadIdx.x >> 5;
  int lane = threadIdx.x & 31, half = lane >> 4, l16 = lane & 15;
  int nt = blockIdx.x * 8 + waveInBlock;          // 256 N-tiles, grid = 32 blocks
  int n = nt * 16 + l16;
  v8f acc0 = {}, acc1 = {}, acc2 = {}, acc3 = {};
  // ---- segment 1: A = xcat (K = 2688) ----
  {
    const float* bp = W_ih + (size_t)n * XK;
    for (int kc = 0; kc < XK; kc += KC) {
      __syncthreads();
      if (waveInBlock == 0) tdm_stage_panel(xcat, XK, kc, sA);
      __syncthreads();
      for (int k = 0; k < KC; k += 4) {
        int kk = k + 2 * half;
        v2f b  = *(const v2f*)(bp + kc + kk);
        v2f a0 = *(const v2f*)(&sA[(size_t)(l16)      * KCP + kk]);
        v2f a1 = *(const v2f*)(&sA[(size_t)(16 + l16) * KCP + kk]);
        v2f a2 = *(const v2f*)(&sA[(size_t)(32 + l16) * KCP + kk]);
        v2f a3 = *(const v2f*)(&sA[(size_t)(48 + l16) * KCP + kk]);
        acc0 = wmma4(a0, b, acc0);
        acc1 = wmma4(a1, b, acc1);
        acc2 = wmma4(a2, b, acc2);
        acc3 = wmma4(a3, b, acc3);
      }
    }
  }
  // ---- segment 2: A = h (K = 1024) ----
  {
    const float* bp = W_hh + (size_t)n * HH;
    for (int kc = 0; kc < HH; kc += KC) {
      __syncthreads();
      if (waveInBlock == 0) tdm_stage_panel(h, HH, kc, sA);
      __syncthreads();
      for (int k = 0; k < KC; k += 4) {
        int kk = k + 2 * half;
        v2f b  = *(const v2f*)(bp + kc + kk);
        v2f a0 = *(const v2f*)(&sA[(size_t)(l16)      * KCP + kk]);
        v2f a1 = *(const v2f*)(&sA[(size_t)(16 + l16) * KCP + kk]);
        v2f a2 = *(const v2f*)(&sA[(size_t)(32 + l16) * KCP + kk]);
        v2f a3 = *(const v2f*)(&sA[(size_t)(48 + l16) * KCP + kk]);
        acc0 = wmma4(a0, b, acc0);
        acc1 = wmma4(a1, b, acc1);
        acc2 = wmma4(a2, b, acc2);
        acc3 = wmma4(a3, b, acc3);
      }
    }
  }
  float bias = b_ih[n] + b_hh[n];
  for (int r = 0; r < 8; ++r) {
    int mr = r + 8 * half;
    gates[(size_t)(mr)      * (4 * HH) + n] = acc0[r] + bias;
    gates[(size_t)(16 + mr) * (4 * HH) + n] = acc1[r] + bias;
    gates[(size_t)(32 + mr) * (4 * HH) + n] = acc2[r] + bias;
    gates[(size_t)(48 + mr) * (4 * HH) + n] = acc3[r] + bias;
  }
}

// LSTM cell elementwise
__global__ void k_lstm_cell(const float* __restrict__ gates, const float* __restrict__ c_in,
                            float* __restrict__ c_out, float* __restrict__ h_out) {
  int idx = blockIdx.x * blockDim.x + threadIdx.x;
  if (idx >= BB * HH) return;
  int b = idx >> 10, j = idx & (HH - 1);
  const float* g = gates + (size_t)b * 4 * HH;
  float gi = g[j], gf = g[j + HH], gg = g[j + 2 * HH], go = g[j + 3 * HH];
  float si = 1.f / (1.f + expf(-gi));
  float sf = 1.f / (1.f + expf(-gf));
  float so = 1.f / (1.f + expf(-go));
  float c2 = sf * c_in[idx] + si * tanhf(gg);
  float h2 = so * tanhf(c2);
  c_out[idx] = c2;
  h_out[idx] = h2;
}

// logits = h2 @ fc_W^T + fc_b -> d_out[b][t][v]   (M=64,N=12000,K=1024)
// Same structure as k_gates: TDM-staged A panel, 4 M-tiles per wave.
__global__ void __launch_bounds__(256) k_logits(
    const float* __restrict__ h, const float* __restrict__ fc_W,
    const float* __restrict__ fc_b, float* __restrict__ outs, int t) {
  __shared__ float sA[BB * KCP];
  const int tilesN = VV / 16;                     // 750
  int waveInBlock = threadIdx.x >> 5;
  int lane = threadIdx.x & 31, half = lane >> 4, l16 = lane & 15;
  int nt = blockIdx.x * 8 + waveInBlock;
  bool active = (nt < tilesN);
  int n = (active ? nt : 0) * 16 + l16;           // inactive waves keep full EXEC, skip store
  v8f acc0 = {}, acc1 = {}, acc2 = {}, acc3 = {};
  const float* bp = fc_W + (size_t)n * HH;
  for (int kc = 0; kc < HH; kc += KC) {
    __syncthreads();
    if (waveInBlock == 0) tdm_stage_panel(h, HH, kc, sA);
    __syncthreads();
    for (int k = 0; k < KC; k += 4) {
      int kk = k + 2 * half;
      v2f b  = *(const v2f*)(bp + kc + kk);
      v2f a0 = *(const v2f*)(&sA[(size_t)(l16)      * KCP + kk]);
      v2f a1 = *(const v2f*)(&sA[(size_t)(16 + l16) * KCP + kk]);
      v2f a2 = *(const v2f*)(&sA[(size_t)(32 + l16) * KCP + kk]);
      v2f a3 = *(const v2f*)(&sA[(size_t)(48 + l16) * KCP + kk]);
      acc0 = wmma4(a0, b, acc0);
      acc1 = wmma4(a1, b, acc1);
      acc2 = wmma4(a2, b, acc2);
      acc3 = wmma4(a3, b, acc3);
    }
  }
  if (!active) return;
  float bias = fc_b[n];
  for (int r = 0; r < 8; ++r) {
    int mr = r + 8 * half;
    outs[((size_t)(mr)      * TT + t) * VV + n] = acc0[r] + bias;
    outs[((size_t)(16 + mr) * TT + t) * VV + n] = acc1[r] + bias;
    outs[((size_t)(32 + mr) * TT + t) * VV + n] = acc2[r] + bias;
    outs[((size_t)(48 + mr) * TT + t) * VV + n] = acc3[r] + bias;
  }
}

// ---------------- launcher ----------------

extern "C" void kernel_launch(void* const* d_in, const int* in_sizes, int n_in,
                              void* d_out, int out_size, void* d_ws, size_t ws_size,
                              hipStream_t stream) {
  (void)in_sizes; (void)n_in; (void)out_size; (void)ws_size;
  const int*   captions = (const int*)  d_in[0];
  const float* features = (const float*)d_in[1];
  const float* cat_enc  = (const float*)d_in[2];
  const float* emb      = (const float*)d_in[3];
  const float* W_ih     = (const float*)d_in[4];
  const float* b_ih     = (const float*)d_in[5];
  const float* W_hh     = (const float*)d_in[6];
  const float* b_hh     = (const float*)d_in[7];
  const float* fc_W     = (const float*)d_in[8];
  const float* fc_b     = (const float*)d_in[9];
  const float* Wh0      = (const float*)d_in[10];
  const float* bh0      = (const float*)d_in[11];
  const float* Wc0      = (const float*)d_in[12];
  const float* bc0      = (const float*)d_in[13];
  const float* att_W    = (const float*)d_in[14];
  const float* att_U    = (const float*)d_in[15];
  const float* att_v    = (const float*)d_in[16];

  float* outs  = (float*)d_out;                       // [B,T,V]
  float* w_out = outs + (size_t)BB * TT * VV;         // [B,T,L]

  float* p = (float*)d_ws;
  float* feat_proj = p; p += (size_t)BB * LL * HH;
  float* meanf     = p; p += (size_t)BB * FF;
  float* embed     = p; p += (size_t)BB * TT * EE;
  float* hA = p; p += (size_t)BB * HH;
  float* cA = p; p += (size_t)BB * HH;
  float* hBuf = p; p += (size_t)BB * HH;
  float* cBuf = p; p += (size_t)BB * HH;
  float* hU   = p; p += (size_t)BB * HH;
  float* ctx  = p; p += (size_t)BB * FF;
  float* xcat = p; p += (size_t)BB * XK;
  float* gates = p; p += (size_t)BB * 4 * HH;

  // ---- setup ----
  k_mean_feat<<<(BB * FF + 255) / 256, 256, 0, stream>>>(features, meanf);
  k_embed<<<(BB * TT * EE + 255) / 256, 256, 0, stream>>>(emb, captions, embed);
  k_init_hc<<<((BB / 16) * (HH / 16) + 7) / 8, 256, 0, stream>>>(meanf, Wh0, bh0, Wc0, bc0, hA, cA);
  k_featproj<<<(((BB * LL) / 16) * (HH / 16) + 7) / 8, 256, 0, stream>>>(features, att_W, feat_proj);

  float *h = hA, *c = cA, *hn = hBuf, *cn = cBuf;
  for (int t = 0; t < TT; ++t) {
    k_hU<<<((BB / 16) * (HH / 16) + 7) / 8, 256, 0, stream>>>(h, att_U, hU);
    k_attention<<<BB, 256, 0, stream>>>(feat_proj, hU, att_v, features, w_out, ctx, t);
    k_xcat<<<(BB * XK + 255) / 256, 256, 0, stream>>>(embed, ctx, cat_enc, xcat, t);
    k_gates<<<(4 * HH / 16) / 8, 256, 0, stream>>>(xcat, W_ih, h, W_hh, b_ih, b_hh, gates);
    k_lstm_cell<<<(BB * HH + 255) / 256, 256, 0, stream>>>(gates, c, cn, hn);
    k_logits<<<((VV / 16) + 7) / 8, 256, 0, stream>>>(hn, fc_W, fc_b, outs, t);
    float* tp;
    tp = h; h = hn; hn = tp;
    tp = c; c = cn; cn = tp;
  }
}